// DPConv_34093450396603
// MI455X (gfx1250) — compile-verified
//
#include <hip/hip_runtime.h>

// ---------------------------------------------------------------------------
// DPConv pipeline for MI455X (gfx1250), fp32 end-to-end via V_WMMA_F32_16X16X4_F32.
// Bandwidth-bound workload (~23 GFLOP vs ~0.6-1 GB @ 23.3 TB/s) => keep fp32
// precision, maximize B-operand (activation) reuse: one wave computes ALL
// output-channel tiles for its 16-pixel tile.
// ---------------------------------------------------------------------------

typedef float v2f __attribute__((ext_vector_type(2)));
typedef float v8f __attribute__((ext_vector_type(8)));

#define HW   10000
#define IMW  100
#define CIN  192
#define CB   64

// D(16x16,f32) = A(16x4,f32) x B(4x16,f32) + C
__device__ __forceinline__ v8f wmma4(v2f a, v2f b, v8f c) {
  return __builtin_amdgcn_wmma_f32_16x16x4_f32(false, a, false, b, (short)0, c,
                                               false, false);
}

// ---------------------------------------------------------------------------
__global__ void __launch_bounds__(256) zero_kernel(float* __restrict__ p, long n) {
  long i = (long)blockIdx.x * blockDim.x + threadIdx.x;
  long stride = (long)gridDim.x * blockDim.x;
  for (; i < n; i += stride) p[i] = 0.0f;
}

// ---------------------------------------------------------------------------
// conv1: xc[n,64,hw] = w1(64x192) @ x[n,192,hw] + b1
// one wave = 16 pixels x ALL 64 out-channels (4 accumulators, B loaded once)
__global__ void __launch_bounds__(256) conv1_kernel(const float* __restrict__ x,
                                                    const float* __restrict__ w1,
                                                    const float* __restrict__ b1,
                                                    float* __restrict__ xc) {
  int wave = (int)((blockIdx.x * blockDim.x + threadIdx.x) >> 5);  // pixel tile
  int lane = threadIdx.x & 31;
  int half = lane >> 4, lid = lane & 15;
  int q  = wave * 16 + lid;
  int n  = q / HW, hw = q % HW;
  const float* xb = x + (size_t)n * CIN * HW + hw;

  v8f zero = {};
  v8f acc[4];
#pragma unroll
  for (int mt = 0; mt < 4; ++mt) acc[mt] = zero;

  for (int kc = 0; kc < CIN; kc += 4) {
    int c0 = kc + half * 2;
    v2f b;
    b.x = xb[(size_t)c0 * HW];
    b.y = xb[(size_t)(c0 + 1) * HW];
#pragma unroll
    for (int mt = 0; mt < 4; ++mt) {
      v2f a;
      a.x = w1[(mt * 16 + lid) * CIN + c0];
      a.y = w1[(mt * 16 + lid) * CIN + c0 + 1];
      acc[mt] = wmma4(a, b, acc[mt]);
    }
  }
  float* ob = xc + (size_t)n * CB * HW + hw;
#pragma unroll
  for (int mt = 0; mt < 4; ++mt)
#pragma unroll
    for (int r = 0; r < 8; ++r) {
      int m = mt * 16 + r + half * 8;
      ob[(size_t)m * HW] = acc[mt][r] + b1[m];
    }
}

// ---------------------------------------------------------------------------
// conv2: z[b,64,kk] = w2(64x64) @ u4 + b2, u4 gathered from xc via window map
__global__ void __launch_bounds__(256) conv2_kernel(const float* __restrict__ xc,
                                                    const float* __restrict__ w2,
                                                    const float* __restrict__ b2,
                                                    float* __restrict__ z,
                                                    int nw, int s, int k) {
  int wave = (int)(((long)blockIdx.x * blockDim.x + threadIdx.x) >> 5);
  int lane = threadIdx.x & 31;
  int half = lane >> 4, lid = lane & 15;
  int kk = k * k;
  long q = (long)wave * 16 + lid;
  int b  = (int)(q / kk);
  int pp = (int)(q % kk);
  int ki = pp / k, kj = pp % k;
  int nn = nw * nw;
  int n  = b / nn, wr = b % nn;
  int wi = wr / nw, wj = wr % nw;
  int off = (wi * s + ki) * IMW + (wj * s + kj);
  const float* xb = xc + (size_t)n * CB * HW + off;

  v8f zero = {};
  v8f acc[4];
#pragma unroll
  for (int mt = 0; mt < 4; ++mt) acc[mt] = zero;

  for (int kc = 0; kc < CB; kc += 4) {
    int c0 = kc + half * 2;
    v2f bb;
    bb.x = xb[(size_t)c0 * HW];
    bb.y = xb[(size_t)(c0 + 1) * HW];
#pragma unroll
    for (int mt = 0; mt < 4; ++mt) {
      v2f a;
      a.x = w2[(mt * 16 + lid) * CB + c0];
      a.y = w2[(mt * 16 + lid) * CB + c0 + 1];
      acc[mt] = wmma4(a, bb, acc[mt]);
    }
  }
  float* zb = z + (size_t)b * CB * kk + pp;
#pragma unroll
  for (int mt = 0; mt < 4; ++mt)
#pragma unroll
    for (int r = 0; r < 8; ++r) {
      int m = mt * 16 + r + half * 8;
      zb[(size_t)m * kk] = acc[mt][r] + b2[m];
    }
}

// ---------------------------------------------------------------------------
// SE: per window b -> channel means -> fc(64->4) relu -> fc(4->64) sigmoid
__global__ void __launch_bounds__(64) se_kernel(const float* __restrict__ z,
                                                const float* __restrict__ wse1,
                                                const float* __restrict__ wse2,
                                                float* __restrict__ sbuf, int kk) {
  int b = blockIdx.x, c = threadIdx.x;
  __shared__ float m[CB];
  __shared__ float t[4];
  const float* zb = z + ((size_t)b * CB + c) * kk;
  float sum = 0.f;
  for (int p = 0; p < kk; ++p) sum += zb[p];
  m[c] = sum / (float)kk;
  __syncthreads();
  if (c < 4) {
    float acc = 0.f;
    for (int j = 0; j < CB; ++j) acc += wse1[c * CB + j] * m[j];
    t[c] = acc > 0.f ? acc : 0.f;
  }
  __syncthreads();
  float acc = 0.f;
#pragma unroll
  for (int j = 0; j < 4; ++j) acc += wse2[c * 4 + j] * t[j];
  sbuf[b * CB + c] = 1.f / (1.f + __expf(-acc));
}

// ---------------------------------------------------------------------------
// analytic overlap count: #{i in [0,nw) : i*s <= h <= i*s + k - 1}
__device__ __forceinline__ int cover_count(int h, int s, int k, int nw) {
  int lo = (h - k + 1) <= 0 ? 0 : (h - k + 1 + s - 1) / s;
  int hi = h / s;
  if (hi > nw - 1) hi = nw - 1;
  return hi - lo + 1;
}

// fold: folded += ( z*SE + dwconv3x3(u4) + bpos ) / count  (atomic overlap-add)
__global__ void __launch_bounds__(256) fold_kernel(const float* __restrict__ z,
                                                   const float* __restrict__ sbuf,
                                                   const float* __restrict__ xc,
                                                   const float* __restrict__ wpos,
                                                   const float* __restrict__ bpos,
                                                   float* __restrict__ folded,
                                                   int nw, int s, int k) {
  int kk = k * k;
  long total = (long)16 * nw * nw * CB * kk;
  long idx = (long)blockIdx.x * blockDim.x + threadIdx.x;
  if (idx >= total) return;
  int pp = (int)(idx % kk);
  long tt = idx / kk;
  int c = (int)(tt % CB);
  int b = (int)(tt / CB);
  int ki = pp / k, kj = pp % k;
  int nn = nw * nw;
  int n = b / nn, wr = b % nn;
  int wi = wr / nw, wj = wr % nw;

  float val = z[idx] * sbuf[b * CB + c];

  const float* xb = xc + ((size_t)n * CB + c) * HW;
  float dw = bpos[c];
#pragma unroll
  for (int di = -1; di <= 1; ++di) {
#pragma unroll
    for (int dj = -1; dj <= 1; ++dj) {
      int kii = ki + di, kjj = kj + dj;
      if (kii >= 0 && kii < k && kjj >= 0 && kjj < k)
        dw += wpos[c * 9 + (di + 1) * 3 + (dj + 1)] *
              xb[(wi * s + kii) * IMW + (wj * s + kjj)];
    }
  }
  int row = wi * s + ki, col = wj * s + kj;
  float rc = 1.f / (float)(cover_count(row, s, k, nw) * cover_count(col, s, k, nw));
  atomicAdd(&folded[((size_t)n * CB + c) * HW + row * IMW + col], (val + dw) * rc);
}

// ---------------------------------------------------------------------------
// conv3: out[n,192,hw] = w3(192x192) @ (concat(folded) + x) + b3
// one wave = 16 pixels x ALL 192 out-channels (12 accumulators);
// the folded/count + x gather is done ONCE per k-step instead of 12x.
__global__ void __launch_bounds__(256) conv3_kernel(const float* __restrict__ x,
                                                    const float* __restrict__ f0,
                                                    const float* __restrict__ f1,
                                                    const float* __restrict__ f2,
                                                    const float* __restrict__ w3,
                                                    const float* __restrict__ b3,
                                                    float* __restrict__ out) {
  int wave = (int)((blockIdx.x * blockDim.x + threadIdx.x) >> 5);  // pixel tile
  int lane = threadIdx.x & 31;
  int half = lane >> 4, lid = lane & 15;
  int q = wave * 16 + lid;
  int n = q / HW, hw = q % HW;
  const float* xb  = x  + (size_t)n * CIN * HW + hw;
  const float* fb0 = f0 + (size_t)n * CB * HW + hw;
  const float* fb1 = f1 + (size_t)n * CB * HW + hw;
  const float* fb2 = f2 + (size_t)n * CB * HW + hw;

  v8f zero = {};
  v8f acc[12];
#pragma unroll
  for (int mt = 0; mt < 12; ++mt) acc[mt] = zero;

#pragma unroll 2
  for (int kc = 0; kc < CIN; kc += 4) {
    int c0 = kc + half * 2;                       // branch uniform per 4-block
    const float* fp = (c0 < CB) ? fb0 : ((c0 < 2 * CB) ? fb1 : fb2);
    int cc = c0 & (CB - 1);
    v2f bb;
    bb.x = fp[(size_t)cc * HW]       + xb[(size_t)c0 * HW];
    bb.y = fp[(size_t)(cc + 1) * HW] + xb[(size_t)(c0 + 1) * HW];
#pragma unroll
    for (int mt = 0; mt < 12; ++mt) {
      v2f a;
      a.x = w3[(mt * 16 + lid) * CIN + c0];
      a.y = w3[(mt * 16 + lid) * CIN + c0 + 1];
      acc[mt] = wmma4(a, bb, acc[mt]);
    }
  }
  float* ob = out + (size_t)n * CIN * HW + hw;
#pragma unroll
  for (int mt = 0; mt < 12; ++mt)
#pragma unroll
    for (int r = 0; r < 8; ++r) {
      int m = mt * 16 + r + half * 8;
      ob[(size_t)m * HW] = acc[mt][r] + b3[m];
    }
}

// ---------------------------------------------------------------------------
extern "C" void kernel_launch(void* const* d_in, const int* in_sizes, int n_in,
                              void* d_out, int out_size, void* d_ws, size_t ws_size,
                              hipStream_t stream) {
  (void)in_sizes; (void)n_in; (void)out_size; (void)ws_size;
  const float* x    = (const float*)d_in[0];
  const float* w1   = (const float*)d_in[1];
  const float* b1   = (const float*)d_in[2];
  const float* w2   = (const float*)d_in[3];
  const float* b2   = (const float*)d_in[4];
  const float* w3   = (const float*)d_in[5];
  const float* b3   = (const float*)d_in[6];
  const float* wpos = (const float*)d_in[7];
  const float* bpos = (const float*)d_in[8];
  const float* wse1 = (const float*)d_in[9];
  const float* wse2 = (const float*)d_in[10];
  float* out = (float*)d_out;
  float* ws  = (float*)d_ws;

  const size_t XC = (size_t)16 * CB * HW;   // 10.24M floats per 64-ch plane set
  float* xc     = ws;                       // [XC]
  float* folded = xc + XC;                  // [3*XC], atomically accumulated
  float* z      = folded + 3 * XC;          // [<= 331776*64] branch conv2 output
  float* sbuf   = z + (size_t)331776 * CB;  // [<= 2304*64] SE scales

  zero_kernel<<<2048, 256, 0, stream>>>(folded, (long)(3 * XC));

  // conv1: 160000/16 = 10000 pixel-tile waves = 1250 blocks (8 waves/block)
  conv1_kernel<<<1250, 256, 0, stream>>>(x, w1, b1, xc);

  const int NWS[3] = {4, 8, 12};
  for (int br = 0; br < 3; ++br) {
    int nw = NWS[br];
    int s = 100 / nw, k = s + 100 % nw, kk = k * k;
    long Ptot = (long)16 * nw * nw * kk;   // 160000 / 262144 / 331776
    long waves = Ptot / 16;                // 10000 / 16384 / 20736 (all %8==0)
    conv2_kernel<<<(int)(waves / 8), 256, 0, stream>>>(xc, w2, b2, z, nw, s, k);
    se_kernel<<<16 * nw * nw, 64, 0, stream>>>(z, wse1, wse2, sbuf, kk);
    long tot = Ptot * CB;
    fold_kernel<<<(int)((tot + 255) / 256), 256, 0, stream>>>(
        z, sbuf, xc, wpos, bpos, folded + (size_t)br * XC, nw, s, k);
  }

  // conv3: 10000 pixel-tile waves = 1250 blocks
  conv3_kernel<<<1250, 256, 0, stream>>>(x, folded, folded + XC, folded + 2 * XC,
                                         w3, b3, out);
}